// GLIPBoxClassification_29025388987109
// MI455X (gfx1250) — compile-verified
//
#include <hip/hip_runtime.h>
#include <hip/hip_bf16.h>

#define IMG_H 900
#define IMG_W 1600
#define NUM_CAMS 6

typedef float v2f __attribute__((ext_vector_type(2)));
typedef float v8f __attribute__((ext_vector_type(8)));

__device__ __forceinline__ float fast_rcp(float x) {
    return __builtin_amdgcn_rcpf(x);   // v_rcp_f32 (TRANS, co-executes with VALU)
}

// Non-negative floats compare identically as unsigned ints -> v_max_u32/v_min_u32,
// avoiding the NaN-canonicalization ops fminf/fmaxf need on LDS-sourced values.
__device__ __forceinline__ float umaxf(float a, float b) {
    return __uint_as_float(max(__float_as_uint(a), __float_as_uint(b)));
}
__device__ __forceinline__ float uminf(float a, float b) {
    return __uint_as_float(min(__float_as_uint(a), __float_as_uint(b)));
}

// ---------------------------------------------------------------------------
// Kernel 1: project 3D boxes into the 6 camera image planes.
// proj:  [6][N][4]  (x1,y1,x2,y2)   valid: [6][N]  (0/1 float)
// ---------------------------------------------------------------------------
__global__ void glip_project_kernel(const float* __restrict__ pred_boxes,
                                    const float* __restrict__ img_aug,   // (1,6,4,4)
                                    const float* __restrict__ lidar_aug, // (1,4,4)
                                    const float* __restrict__ l2i,       // (1,6,4,4)
                                    float* __restrict__ proj,
                                    float* __restrict__ validO,
                                    int N)
{
    int n = blockIdx.x * blockDim.x + threadIdx.x;
    if (n >= N) return;

    float cx = pred_boxes[n*7+0], cy = pred_boxes[n*7+1], cz = pred_boxes[n*7+2];
    float dx = pred_boxes[n*7+3], dy = pred_boxes[n*7+4], dz = pred_boxes[n*7+5];
    float yaw = pred_boxes[n*7+6];
    float cs = __cosf(yaw), sn = __sinf(yaw);

    // lidar_aug rotation inverse (adjugate) + translation
    float a00 = lidar_aug[0], a01 = lidar_aug[1], a02 = lidar_aug[2],  tx = lidar_aug[3];
    float a10 = lidar_aug[4], a11 = lidar_aug[5], a12 = lidar_aug[6],  ty = lidar_aug[7];
    float a20 = lidar_aug[8], a21 = lidar_aug[9], a22 = lidar_aug[10], tz = lidar_aug[11];
    float det = a00*(a11*a22 - a12*a21) - a01*(a10*a22 - a12*a20) + a02*(a10*a21 - a11*a20);
    float id  = fast_rcp(det);
    float i00 =  (a11*a22 - a12*a21)*id, i01 = -(a01*a22 - a02*a21)*id, i02 =  (a01*a12 - a02*a11)*id;
    float i10 = -(a10*a22 - a12*a20)*id, i11 =  (a00*a22 - a02*a20)*id, i12 = -(a00*a12 - a02*a10)*id;
    float i20 =  (a10*a21 - a11*a20)*id, i21 = -(a00*a21 - a01*a20)*id, i22 =  (a00*a11 - a01*a10)*id;

    const float sxl[8] = { 0.5f, 0.5f,-0.5f,-0.5f, 0.5f, 0.5f,-0.5f,-0.5f};
    const float syl[8] = { 0.5f,-0.5f,-0.5f, 0.5f, 0.5f,-0.5f,-0.5f, 0.5f};
    const float szl[8] = {-0.5f,-0.5f,-0.5f,-0.5f, 0.5f, 0.5f, 0.5f, 0.5f};

    float ptx[8], pty[8], ptz[8];
#pragma unroll
    for (int i = 0; i < 8; ++i) {
        float lx = dx * sxl[i], ly = dy * syl[i], lz = dz * szl[i];
        // rotated = corner @ rot, rot rows = [c,s,0],[-s,c,0],[0,0,1]
        float rx = lx*cs - ly*sn;
        float ry = lx*sn + ly*cs;
        float rz = lz;
        float wx = rx + cx - tx, wy = ry + cy - ty, wz = rz + cz - tz;
        ptx[i] = i00*wx + i01*wy + i02*wz;
        pty[i] = i10*wx + i11*wy + i12*wz;
        ptz[i] = i20*wx + i21*wy + i22*wz;
    }

#pragma unroll
    for (int c = 0; c < NUM_CAMS; ++c) {
        const float* L  = l2i     + c*16;
        const float* IA = img_aug + c*16;
        float mnx = 3.0e38f, mny = 3.0e38f, mxx = -3.0e38f, mxy = -3.0e38f;
#pragma unroll
        for (int i = 0; i < 8; ++i) {
            float X = L[0]*ptx[i] + L[1]*pty[i] + L[2]*ptz[i]  + L[3];
            float Y = L[4]*ptx[i] + L[5]*pty[i] + L[6]*ptz[i]  + L[7];
            float Z = L[8]*ptx[i] + L[9]*pty[i] + L[10]*ptz[i] + L[11];
            float zc = fminf(fmaxf(Z, 1e-5f), 100000.0f);
            float zr = fast_rcp(zc);
            float x = X * zr, y = Y * zr;
            float qx = IA[0]*x + IA[1]*y + IA[2]*zc  + IA[3];
            float qy = IA[4]*x + IA[5]*y + IA[6]*zc  + IA[7];
            qx = fminf(fmaxf(qx, -2.0e9f), 2.0e9f);
            qy = fminf(fmaxf(qy, -2.0e9f), 2.0e9f);
            int ipx = (int)qx;          // truncate toward zero, as astype(int32)
            int ipy = (int)qy;
            ipx = min(max(ipx, 0), IMG_W);
            ipy = min(max(ipy, 0), IMG_H);
            float fx = (float)ipx, fy = (float)ipy;
            mnx = fminf(mnx, fx); mny = fminf(mny, fy);
            mxx = fmaxf(mxx, fx); mxy = fmaxf(mxy, fy);
        }
        float v = ((mxx - mnx) > 0.0f && (mxy - mny) > 0.0f) ? 1.0f : 0.0f;
        proj[(c*N + n)*4 + 0] = mnx;
        proj[(c*N + n)*4 + 1] = mny;
        proj[(c*N + n)*4 + 2] = mxx;
        proj[(c*N + n)*4 + 3] = mxy;
        validO[c*N + n] = v;
    }
}

// ---------------------------------------------------------------------------
// Kernel 2: fused IoU + (N x Mseg)·(Mseg x 16) partial reduction via
// V_WMMA_F32_16X16X4_F32.  Grid = (N/128, S): blockIdx.y selects an M-segment.
// Masking strategy (all folded out of the inner loop):
//   - wrong-batch / bad-cam detections get a zero label row (B operand = 0)
//   - invalid projections are stored as the degenerate box (0,0,0,0), whose
//     IoU with any non-negative det box is exactly 0 — no valid multiply.
// ---------------------------------------------------------------------------
#define MCHUNK 128
#define MSEG   768          // detections per segment (multiple of MCHUNK)

__global__ void __launch_bounds__(256)
glip_wmma_kernel(const float* __restrict__ det_boxes,   // [M][4]
                 const int*   __restrict__ det_labels,  // [M]
                 const float* __restrict__ det_scores,  // [M]
                 const int*   __restrict__ det_batch,   // [M]
                 const int*   __restrict__ det_cam,     // [M]
                 const float* __restrict__ proj,        // [6][N][4]
                 const float* __restrict__ validA,      // [6][N]
                 float* __restrict__ partial,           // [S][N][16]
                 int N, int M)
{
    __shared__ float  lmS[MCHUNK * 16];                // scored one-hot rows (16-wide)
    __shared__ float4 boxS[MCHUNK];                    // det boxes (b128 loads)
    __shared__ int    camS[MCHUNK];                    // cam * 128 (pre-scaled f4 index)
    __shared__ float4 projS[NUM_CAMS * 8 * 16];        // [cam][wave][row], valid folded

    const int t    = threadIdx.x;
    const int lane = t & 31;
    const int w    = t >> 5;          // wave id 0..7
    const int r    = lane & 15;       // row within 16-box tile / class column
    const int half = lane >> 4;       // 0: K=0,1   1: K=2,3  (f32 16x4 A layout)
    const int kb   = half * 2;
    const int wr   = w * 16 + r;      // per-lane float4 sub-index into projS
    const int nbase = blockIdx.x * 128 + w * 16;

    const int m_begin = blockIdx.y * MSEG;
    const int m_end   = min(M, m_begin + MSEG);

    // Stage this wave's 6-cam projected boxes into LDS; zero out invalid ones.
    {
        int n = nbase + r;
#pragma unroll
        for (int cc = 0; cc < 3; ++cc) {
            int c = half * 3 + cc;
            float v = validA[c*N + n];
            float4 p;
            p.x = proj[(c*N + n)*4 + 0] * v;
            p.y = proj[(c*N + n)*4 + 1] * v;
            p.z = proj[(c*N + n)*4 + 2] * v;
            p.w = proj[(c*N + n)*4 + 3] * v;
            projS[c * 128 + wr] = p;
        }
    }

    v8f acc = {};

    for (int m0 = m_begin; m0 < m_end; m0 += MCHUNK) {
        __syncthreads();   // previous chunk fully consumed (also covers projS 1st use)
        if (t < MCHUNK) {
            int m = m0 + t;
            float s = 0.0f; int cls = -1, cam = 0;
            float4 bb = {0.f, 0.f, 0.f, 0.f};
            if (m < m_end) {
                s   = det_scores[m];
                cls = det_labels[m] - 1;
                int bi = det_batch[m];
                cam = det_cam[m];
                if (bi != 0 || cam < 0 || cam >= NUM_CAMS) { cam = 0; s = 0.0f; }
                bb.x = det_boxes[m*4+0]; bb.y = det_boxes[m*4+1];
                bb.z = det_boxes[m*4+2]; bb.w = det_boxes[m*4+3];
                if (m0 + MCHUNK < m_end)
                    __builtin_prefetch(det_boxes + (m0 + MCHUNK + t) * 4, 0, 0);
            }
#pragma unroll
            for (int q = 0; q < 16; ++q)
                lmS[t*16 + q] = (q == cls) ? s : 0.0f;
            boxS[t] = bb;
            camS[t] = cam * 128;
        }
        __syncthreads();

        // 32 WMMA steps: each consumes 4 detections for this wave's 16 boxes.
#pragma unroll 8
        for (int kk = 0; kk < MCHUNK; kk += 4) {
            float a[2];
#pragma unroll
            for (int j = 0; j < 2; ++j) {
                int  mloc = kk + kb + j;
                float4 pb = projS[camS[mloc] + wr];   // single ds_load_b128
                float4 ab = boxS[mloc];               // single ds_load_b128
                float ltx = umaxf(ab.x, pb.x), lty = umaxf(ab.y, pb.y);
                float rbx = uminf(ab.z, pb.z), rby = uminf(ab.w, pb.w);
                float iw  = fmaxf(rbx - ltx, 0.0f);
                float ih  = fmaxf(rby - lty, 0.0f);
                float inter = iw * ih;
                float areaA = (ab.z - ab.x) * (ab.w - ab.y);
                float areaB = (pb.z - pb.x) * (pb.w - pb.y);
                float uni   = areaA + areaB - inter;
                a[j] = (uni > 0.0f) ? (inter * fast_rcp(uni)) : 0.0f;
            }
            v2f A = { a[0], a[1] };
            v2f B = { lmS[(kk + kb)     * 16 + r],
                      lmS[(kk + kb + 1) * 16 + r] };
            // D = A(16x4,f32) * B(4x16,f32) + C : V_WMMA_F32_16X16X4_F32
            acc = __builtin_amdgcn_wmma_f32_16x16x4_f32(
                      false, A, false, B, (short)0, acc, false, false);
        }
    }

    // C layout: VGPR v -> row (v + half*8), col = lane&15.  Raw partial sums.
    float* seg = partial + (size_t)blockIdx.y * N * 16;
#pragma unroll
    for (int v = 0; v < 8; ++v) {
        int row = v + half * 8;
        seg[(size_t)(nbase + row) * 16 + r] = acc[v];
    }
}

// ---------------------------------------------------------------------------
// Kernel 3: sum the S partial slices, apply 1/(1e-5 + sum valid), max/argmax.
// ---------------------------------------------------------------------------
__global__ void glip_reduce_kernel(const float* __restrict__ partial, // [S][N][16]
                                   const float* __restrict__ validA,  // [6][N]
                                   float* __restrict__ out_labels,
                                   float* __restrict__ out_scores,
                                   int N, int S)
{
    int n = blockIdx.x * blockDim.x + threadIdx.x;
    if (n >= N) return;

    float d = 1e-5f;
#pragma unroll
    for (int c = 0; c < NUM_CAMS; ++c)
        d += validA[c*N + n];
    float rd = fast_rcp(d);

    float sums[10];
#pragma unroll
    for (int q = 0; q < 10; ++q) sums[q] = 0.0f;
    for (int s = 0; s < S; ++s) {
        const float* p = partial + ((size_t)s * N + n) * 16;
#pragma unroll
        for (int q = 0; q < 10; ++q) sums[q] += p[q];
    }

    float best = sums[0] * rd;
    int   bi   = 0;
#pragma unroll
    for (int q = 1; q < 10; ++q) {
        float vq = sums[q] * rd;
        if (vq > best) { best = vq; bi = q; }
    }
    out_labels[n] = (float)(bi + 1);
    out_scores[n] = best;
}

// ---------------------------------------------------------------------------
extern "C" void kernel_launch(void* const* d_in, const int* in_sizes, int n_in,
                              void* d_out, int out_size, void* d_ws, size_t ws_size,
                              hipStream_t stream) {
    const float* pred_boxes = (const float*)d_in[0];
    const float* img_aug    = (const float*)d_in[1];
    const float* lidar_aug  = (const float*)d_in[2];
    const float* l2i        = (const float*)d_in[3];
    const float* det_boxes  = (const float*)d_in[4];
    const int*   det_labels = (const int*)  d_in[5];
    const float* det_scores = (const float*)d_in[6];
    const int*   det_batch  = (const int*)  d_in[7];
    const int*   det_cam    = (const int*)  d_in[8];

    const int N = in_sizes[0] / 7;   // 4096
    const int M = in_sizes[4] / 4;   // 6144
    const int S = (M + MSEG - 1) / MSEG;   // 8 segments for M=6144

    float* proj    = (float*)d_ws;               // 6*N*4 floats
    float* valid   = proj + 6 * N * 4;           // 6*N floats
    float* partial = valid + 6 * N;              // S*N*16 floats

    float* out_labels = (float*)d_out;           // first N entries
    float* out_scores = out_labels + N;          // next  N entries

    glip_project_kernel<<<(N + 255) / 256, 256, 0, stream>>>(
        pred_boxes, img_aug, lidar_aug, l2i, proj, valid, N);

    dim3 grid2(N / 128, S);
    glip_wmma_kernel<<<grid2, 256, 0, stream>>>(
        det_boxes, det_labels, det_scores, det_batch, det_cam,
        proj, valid, partial, N, M);

    glip_reduce_kernel<<<(N + 255) / 256, 256, 0, stream>>>(
        partial, valid, out_labels, out_scores, N, S);
}